// RoIPoolMa_29137058136181
// MI455X (gfx1250) — compile-verified
//
#include <hip/hip_runtime.h>
#include <stdint.h>

// RoI max-pooling for MI455X (gfx1250).
//   feature: [C=256, H=64, W=64] f32, rois: [256,4] f32 -> out: [256,256,7,7] f32
//
// Strategy: one workgroup per ROI. The ROI's feature window (<=9x9 cells, all
// 256 channels) is a 3-D strided tile -> a single Tensor Data Mover
// tensor_load_to_lds DMA stages it into LDS (~83 KB), then 8 wave32s compute
// the 49x256 bin maxima from LDS with coalesced global stores.

#define PH_ 7
#define PW_ 7
#define C_  256
#define H_  64
#define W_  64
#define NROI_ 256
#define MAXR_ 10          // max ROI extent on the 64x64 map (120px/16 -> <=9; +1 margin)
#define SCALE_ 0.0625f

typedef __attribute__((ext_vector_type(4))) unsigned int uint32x4_t_;
typedef __attribute__((ext_vector_type(8))) int          int32x8_t_;
typedef __attribute__((ext_vector_type(4))) int          int32x4_t_;

__device__ __forceinline__ int clampi(int v, int lo, int hi) {
    return v < lo ? lo : (v > hi ? hi : v);
}

__device__ __forceinline__ unsigned int rfl_u32(unsigned int v) {
    return static_cast<unsigned int>(__builtin_amdgcn_readfirstlane(static_cast<int>(v)));
}
__device__ __forceinline__ int rfl_i32(int v) {
    return __builtin_amdgcn_readfirstlane(v);
}

extern "C" __global__ __launch_bounds__(256)
void roipool_tdm_kernel(const float* __restrict__ feature,
                        const float* __restrict__ rois,
                        float* __restrict__ out)
{
    // TDM fills LDS contiguously: x fastest, then y, then z(channel)
    // -> layout tile[c][y][x] with extents [C_][regH][regW]
    extern __shared__ float tile[];

    const int n   = blockIdx.x;
    const int tid = threadIdx.x;

    // --- Block-uniform ROI geometry (rintf == round-half-even == jnp.round) ---
    const int x1 = (int)rintf(rois[n * 4 + 0] * SCALE_);
    const int y1 = (int)rintf(rois[n * 4 + 1] * SCALE_);
    const int x2 = (int)rintf(rois[n * 4 + 2] * SCALE_);
    const int y2 = (int)rintf(rois[n * 4 + 3] * SCALE_);
    const float bin_h = (float)max(y2 - y1 + 1, 1) / (float)PH_;
    const float bin_w = (float)max(x2 - x1 + 1, 1) / (float)PW_;

    // Union of all clipped bins: rows [r0,r1), cols [c0,c1)
    const int r0 = clampi(y1, 0, H_), r1 = clampi(y2 + 1, 0, H_);
    const int c0 = clampi(x1, 0, W_), c1 = clampi(x2 + 1, 0, W_);
    int regH = r1 - r0, regW = c1 - c0;
    if (regH > MAXR_) regH = MAXR_;   // never triggers for the stated input ranges
    if (regW > MAXR_) regW = MAXR_;
    const bool nonempty = (regH > 0) && (regW > 0);

    if (nonempty) {
#if defined(__gfx1250__) && __has_builtin(__builtin_amdgcn_tensor_load_to_lds)
        if (tid < 32) {   // wave 0 drives the Tensor Data Mover (EXEC ignored by TDM)
            const uint64_t gaddr = (uint64_t)(uintptr_t)(const void*)
                                   (feature + (r0 * W_ + c0));      // tile start
            const uint32_t ldsa  = (uint32_t)(uintptr_t)(void*)tile; // LDS byte offset

            // ---- D# group 0 (128b): count=1 | lds_addr | global_addr | type=2 ----
            uint32_t g0w0 = 1u;                                     // count=1, user mode
            uint32_t g0w1 = ldsa;
            uint32_t g0w2 = (uint32_t)gaddr;                        // global_addr[31:0]
            uint32_t g0w3 = (uint32_t)((gaddr >> 32) & 0x01FFFFFFu) // global_addr[56:32]
                          | (2u << 30);                             // type=2 ("image")

            // ---- D# group 1 (256b) ----
            // w0: wg_mask=0 | data_size=2 (4B) | no barrier/iterate/pad
            int g1w0 = (2 << 16);
            // tensor_dim0 = W (bits 79:48), tensor_dim1 = H (bits 111:80)
            int g1w1 = (W_ & 0xFFFF) << 16;                         // dim0 lo16
            int g1w2 = ((W_ >> 16) & 0xFFFF) | ((H_ & 0xFFFF) << 16);
            // tile_dim0 = regW (bits 127:112)
            int g1w3 = ((H_ >> 16) & 0xFFFF) | ((regW & 0xFFFF) << 16);
            // tile_dim1 = regH, tile_dim2 = C (all channels in one DMA)
            int g1w4 = (regH & 0xFFFF) | ((C_ & 0xFFFF) << 16);
            // tensor_dim0_stride = W (row stride), tensor_dim1_stride = H*W (channel)
            int g1w5 = W_;                                          // stride0[31:0]
            int g1w6 = (((H_ * W_) & 0xFFFF) << 16);                // stride0 hi | stride1 lo16
            int g1w7 = (H_ * W_) >> 16;                             // stride1[47:16]

            // ---- D# group 2: tensor_dim2 = C; dim3/tile_dim3 unused ----
            int g2w0 = C_;

            // Force SGPR-uniform descriptor operands.
            uint32x4_t_ g0 = { rfl_u32(g0w0), rfl_u32(g0w1),
                               rfl_u32(g0w2), rfl_u32(g0w3) };
            int32x8_t_  g1 = { rfl_i32(g1w0), rfl_i32(g1w1), rfl_i32(g1w2),
                               rfl_i32(g1w3), rfl_i32(g1w4), rfl_i32(g1w5),
                               rfl_i32(g1w6), rfl_i32(g1w7) };
            int32x4_t_  g2 = { rfl_i32(g2w0), 0, 0, 0 };
            int32x4_t_  g3 = { 0, 0, 0, 0 };
            int32x8_t_  g4 = { 0, 0, 0, 0, 0, 0, 0, 0 };  // no 5th SGPR group in ISA; zeros

            // clang-23 / therock-10.0 form: 6 args
            __builtin_amdgcn_tensor_load_to_lds(g0, g1, g2, g3, g4, /*cpol=*/0);
            __builtin_amdgcn_s_wait_tensorcnt(0);
        }
#else
        // Fallback: cooperative staged load (no TDM builtin on this toolchain)
        const int per_c0 = regW * regH;
        for (int i = tid; i < per_c0 * C_; i += 256) {
            int c = i / per_c0;
            int r = i - c * per_c0;
            int y = r / regW, x = r - y * regW;
            tile[i] = feature[c * (H_ * W_) + (r0 + y) * W_ + (c0 + x)];
        }
#endif
    }
    __syncthreads();   // TDM data visible to all 8 waves after the barrier

    // --- Each thread produces 49 outputs; consecutive tids -> coalesced stores ---
    const int per_c = regH * regW;
    for (int idx = tid; idx < C_ * PH_ * PW_; idx += 256) {
        const int c  = idx / (PH_ * PW_);
        const int b  = idx - c * (PH_ * PW_);
        const int ph = b / PW_;
        const int pw = b - ph * PW_;

        // Bin edges exactly as the reference (fp32 floor/ceil, clip to [0,64])
        int hs = clampi((int)floorf((float)ph * bin_h) + y1, 0, H_);
        int he = clampi((int)ceilf((float)(ph + 1) * bin_h) + y1, 0, H_);
        int ws = clampi((int)floorf((float)pw * bin_w) + x1, 0, W_);
        int we = clampi((int)ceilf((float)(pw + 1) * bin_w) + x1, 0, W_);
        // Defensive: never read past the staged window
        he = min(he, r0 + regH);
        we = min(we, c0 + regW);

        float m = 0.0f;    // empty bins -> 0 (reference maps -inf to 0)
        if (nonempty && he > hs && we > ws) {
            float acc = -INFINITY;
            const float* src = tile + c * per_c;
            for (int h = hs; h < he; ++h) {
                const int rowoff = (h - r0) * regW - c0;
                for (int w = ws; w < we; ++w)
                    acc = fmaxf(acc, src[rowoff + w]);
            }
            m = acc;
        }
        out[(n * C_ + c) * (PH_ * PW_) + b] = m;
    }
}

extern "C" void kernel_launch(void* const* d_in, const int* in_sizes, int n_in,
                              void* d_out, int out_size, void* d_ws, size_t ws_size,
                              hipStream_t stream) {
    const float* feature = (const float*)d_in[0];   // [256,64,64] f32
    const float* rois    = (const float*)d_in[1];   // [256,4] f32
    float* out           = (float*)d_out;           // [256,256,7,7] f32

    const size_t shmem = (size_t)MAXR_ * MAXR_ * C_ * sizeof(float); // 102,400 B (<320KB/WGP)
    roipool_tdm_kernel<<<NROI_, 256, shmem, stream>>>(feature, rois, out);
}